// CollectAttention_50199577755805
// MI455X (gfx1250) — compile-verified
//
#include <hip/hip_runtime.h>

// PSA 'collect' gather, N=2, H=W=64, C=127^2=16129.
// out[n, i*64+j, h, w] = x[n, (i-h+63)*127 + (j-w+63), h, w]
//
// Pure data movement (0 FLOPs, 134 MB in + 134 MB out). Entirely on the CDNA5
// async data-mover path; no data ever passes through VGPRs.
//
// Phase 1: the 127 partial band rows are folded into 64 FULL virtual rows by
//   pairing complementary rows (b, b+64) whose w-ranges tile [0,64) exactly.
//   => 16 unconditional, fully-occupied global_load_async_to_lds_b32 per
//   thread, LDS destination applies the shear (element lands at its final
//   output-tile position). No predication, no masked lanes.
// Phase 2: LDS tile is the finished output tile -> 4x
//   global_store_async_from_lds_b128 per thread (16B/lane, linear copy).

#define HW_ 4096u        // H*W
#define C_  16129u       // (2W-1)^2

__global__ __launch_bounds__(256) void psa_collect_async(const float* __restrict__ x,
                                                         float* __restrict__ out) {
    __shared__ float tile[64 * 64];   // finished 64x64 output tile, 16 KB

    const unsigned tid = threadIdx.x;
    const unsigned w   = tid & 63u;   // column
    const unsigned jg  = tid >> 6;    // 0..3: virtual-row group

    const unsigned bid = blockIdx.x;  // = ((n*64 + h)*64 + i)
    const unsigned i = bid & 63u;
    const unsigned h = (bid >> 6) & 63u;
    const unsigned n = bid >> 12;

    const unsigned a = i + 63u - h;   // 0..126, fixed per block

    const unsigned ldsbase = (unsigned)(unsigned long long)&tile[0]; // LDS byte addr

    // ---- Phase 1: pair-packed async gather x -> LDS (shear at LDS dst) ----
    // Virtual row v = kb*4 + jg, u = v + w + 1:
    //   output row j = u & 63          -> lds byte = ldsbase + w*4 + (j << 8)
    //   channel row b = v + 64-(u&64)  -> glb byte = xw0 + kb*65536 - ((u&64)<<14)
    const unsigned lw0 = ldsbase + w * 4u;
    const unsigned xw0 = ((n * C_ + a * 127u) * HW_ + h * 64u + w) * 4u
                       + jg * 16384u + 0x100000u;   // includes v*16384 (jg part) + 64*16384

    unsigned u = jg + w + 1u;
#pragma unroll
    for (unsigned kb = 0; kb < 16u; ++kb) {
        const unsigned loff = lw0 + ((u & 63u) << 8);
        const unsigned goff = xw0 + kb * 65536u - ((u & 64u) << 14);
        asm volatile("global_load_async_to_lds_b32 %0, %1, %2"
                     :
                     : "v"(loff), "v"(goff), "s"(x)
                     : "memory");
        u += 4u;
    }
    asm volatile("s_wait_asynccnt 0x0" ::: "memory");
    __syncthreads();

    // ---- Phase 2: vector async scatter LDS -> out (16B per lane) ----
    // out row j: byte offset ((n*HW + i*64 + j)*HW + h*64)*4, 256 B per row.
    const unsigned obyte0 = ((n * HW_ + i * 64u) * HW_ + h * 64u) * 4u;

#pragma unroll
    for (unsigned k = 0; k < 4u; ++k) {
        const unsigned cidx = k * 256u + tid;            // 16B chunk id, 0..1023
        const unsigned j    = cidx >> 4;                 // output row 0..63
        const unsigned loff = ldsbase + cidx * 16u;      // contiguous in LDS
        const unsigned goff = obyte0 + j * (HW_ * 4u) + (cidx & 15u) * 16u;
        asm volatile("global_store_async_from_lds_b128 %0, %1, %2"
                     :
                     : "v"(goff), "v"(loff), "s"(out)
                     : "memory");
    }
    // S_ENDPGM implies wait-idle, but make the dependency explicit:
    asm volatile("s_wait_asynccnt 0x0" ::: "memory");
}

extern "C" void kernel_launch(void* const* d_in, const int* in_sizes, int n_in,
                              void* d_out, int out_size, void* d_ws, size_t ws_size,
                              hipStream_t stream) {
    (void)in_sizes; (void)n_in; (void)out_size; (void)d_ws; (void)ws_size;
    const float* x  = (const float*)d_in[0];
    float* out      = (float*)d_out;

    // One workgroup per (n, h, i): 2*64*64 = 8192 blocks, 256 threads (8 waves).
    dim3 grid(8192);
    dim3 block(256);
    psa_collect_async<<<grid, block, 0, stream>>>(x, out);
}